// RandomFeatureAttention_738734375520
// MI455X (gfx1250) — compile-verified
//
#include <hip/hip_runtime.h>

typedef __attribute__((ext_vector_type(2))) float v2f;
typedef __attribute__((ext_vector_type(8))) float v8f;

namespace {
constexpr int kH = 16;
constexpr int kBH = 64;          // B*H
constexpr int kN = 4096;
constexpr int kD = 64;           // head dim == projection dim e
constexpr int kE2 = 128;         // 2e feature dim
constexpr int kNTile = 128;      // rows of n processed per LDS tile
constexpr int kNSplit = 16;      // N-chunks for kv kernel
constexpr int kChunk = kN / kNSplit;          // 256
constexpr int kTilesPerChunk = kChunk / kNTile; // 2
}

__device__ __forceinline__ v8f wmma_f32(v2f a, v2f b, v8f c) {
  // D = A(16x4,f32) * B(4x16,f32) + C(16x16,f32)
  return __builtin_amdgcn_wmma_f32_16x16x4_f32(false, a, false, b,
                                               (short)0, c, false, false);
}

__global__ void rfa_zero_kernel(float* __restrict__ p, int n) {
  int i = blockIdx.x * blockDim.x + threadIdx.x;
  int stride = gridDim.x * blockDim.x;
  for (; i < n; i += stride) p[i] = 0.0f;
}

// ---------------------------------------------------------------------------
// Kernel A: kv[bh][i][j] += sum_n phi_k[n][i] * v[n][j]   (atomic reduction)
// block = (bh, chunk), 256 threads = 8 waves
// ---------------------------------------------------------------------------
__global__ __launch_bounds__(256)
void rfa_kv_kernel(const float* __restrict__ kmat, const float* __restrict__ vmat,
                   const float* __restrict__ proj, float* __restrict__ kv_out) {
  __shared__ float s_proj[kD][kD + 1];        // 64 x 65
  __shared__ float s_k[kNTile][kD + 1];       // 128 x 65
  __shared__ float s_v[kNTile][kD + 1];       // 128 x 65
  __shared__ float s_phi[kNTile][kE2 + 1];    // 128 x 129

  const int bh = blockIdx.x;
  const int chunk = blockIdx.y;
  const int h = bh % kH;
  const int tid = threadIdx.x;
  const int lane = tid & 31;
  const int wave = tid >> 5;
  const int m = lane & 15;            // row (A) / col (B,D) within 16x16 tile
  const int kk = (lane >> 4) * 2;     // K sub-offset for A/B fragments

  // projections[h] : P[d][e], 4096 floats
  for (int i = tid; i < kD * kD; i += 256) {
    s_proj[i / kD][i % kD] = proj[h * kD * kD + i];
  }

  v8f acc_kv[4];
  for (int t = 0; t < 4; ++t) acc_kv[t] = {};

  const float* kbase = kmat + (size_t)bh * kN * kD + (size_t)chunk * kChunk * kD;
  const float* vbase = vmat + (size_t)bh * kN * kD + (size_t)chunk * kChunk * kD;

  for (int nt = 0; nt < kTilesPerChunk; ++nt) {
    __syncthreads();
    // cooperative load of 128x64 k and v tiles (float4)
    for (int i = tid; i < kNTile * kD / 4; i += 256) {
      const int row = i / (kD / 4);
      const int c4 = (i % (kD / 4)) * 4;
      const float4 kv4 = *(const float4*)(kbase + (size_t)nt * kNTile * kD + row * kD + c4);
      const float4 vv4 = *(const float4*)(vbase + (size_t)nt * kNTile * kD + row * kD + c4);
      s_k[row][c4 + 0] = kv4.x; s_k[row][c4 + 1] = kv4.y;
      s_k[row][c4 + 2] = kv4.z; s_k[row][c4 + 3] = kv4.w;
      s_v[row][c4 + 0] = vv4.x; s_v[row][c4 + 1] = vv4.y;
      s_v[row][c4 + 2] = vv4.z; s_v[row][c4 + 3] = vv4.w;
    }
    __syncthreads();

    // kp = k_tile(128x64) x P(64x64); wave w owns row tile i0 = 16w
    {
      const int i0 = wave * 16;
      for (int tj = 0; tj < 4; ++tj) {
        const int j0 = tj * 16;
        v8f acc = {};
        for (int kb = 0; kb < kD; kb += 4) {
          v2f a, b;
          a.x = s_k[i0 + m][kb + kk];
          a.y = s_k[i0 + m][kb + kk + 1];
          b.x = s_proj[kb + kk][j0 + m];
          b.y = s_proj[kb + kk + 1][j0 + m];
          acc = wmma_f32(a, b, acc);
        }
        // phi_k = [sin(kp), cos(kp)] / sqrt(64)
        for (int g = 0; g < 8; ++g) {
          const int row = i0 + g + (lane >> 4) * 8;
          const int col = j0 + m;
          const float val = acc[g];
          s_phi[row][col] = __sinf(val) * 0.125f;
          s_phi[row][col + kD] = __cosf(val) * 0.125f;
        }
      }
    }
    __syncthreads();

    // kv(128x64) += phi_k^T(128x128n) x v(128n x 64); wave w owns kv rows 16w..
    {
      const int i0 = wave * 16;
      for (int tj = 0; tj < 4; ++tj) {
        const int j0 = tj * 16;
        v8f acc = acc_kv[tj];
        for (int kb = 0; kb < kNTile; kb += 4) {
          v2f a, b;
          a.x = s_phi[kb + kk][i0 + m];       // transposed read (conflict-free)
          a.y = s_phi[kb + kk + 1][i0 + m];
          b.x = s_v[kb + kk][j0 + m];
          b.y = s_v[kb + kk + 1][j0 + m];
          acc = wmma_f32(a, b, acc);
        }
        acc_kv[tj] = acc;
      }
    }
  }

  // reduce partial kv into global workspace
  {
    const int i0 = wave * 16;
    float* out = kv_out + (size_t)bh * kE2 * kD;
    for (int tj = 0; tj < 4; ++tj) {
      const int j0 = tj * 16;
      for (int g = 0; g < 8; ++g) {
        const int row = i0 + g + (lane >> 4) * 8;
        const int col = j0 + m;
        atomicAdd(&out[row * kD + col], acc_kv[tj][g]);
      }
    }
  }
}

// ---------------------------------------------------------------------------
// Kernel B: out[n][d] = sum_e phi_q[n][e] * kv[e][d]
// block = (bh, n-tile of 128 rows), 256 threads = 8 waves
// ---------------------------------------------------------------------------
__global__ __launch_bounds__(256)
void rfa_out_kernel(const float* __restrict__ qmat, const float* __restrict__ proj,
                    const float* __restrict__ kv_in, float* __restrict__ out) {
  __shared__ float s_proj[kD][kD + 1];
  __shared__ float s_q[kNTile][kD + 1];
  __shared__ float s_phi[kNTile][kE2 + 1];
  __shared__ float s_kv[kE2][kD + 1];

  const int bh = blockIdx.x;
  const int tile = blockIdx.y;
  const int h = bh % kH;
  const int tid = threadIdx.x;
  const int lane = tid & 31;
  const int wave = tid >> 5;
  const int m = lane & 15;
  const int kk = (lane >> 4) * 2;

  for (int i = tid; i < kD * kD; i += 256) {
    s_proj[i / kD][i % kD] = proj[h * kD * kD + i];
  }
  for (int i = tid; i < kE2 * kD; i += 256) {
    s_kv[i / kD][i % kD] = kv_in[(size_t)bh * kE2 * kD + i];
  }
  const float* qbase = qmat + (size_t)bh * kN * kD + (size_t)tile * kNTile * kD;
  for (int i = tid; i < kNTile * kD / 4; i += 256) {
    const int row = i / (kD / 4);
    const int c4 = (i % (kD / 4)) * 4;
    const float4 q4 = *(const float4*)(qbase + row * kD + c4);
    s_q[row][c4 + 0] = q4.x; s_q[row][c4 + 1] = q4.y;
    s_q[row][c4 + 2] = q4.z; s_q[row][c4 + 3] = q4.w;
  }
  __syncthreads();

  // qp = q_tile x P ; phi_q = [sin, cos]/8
  {
    const int i0 = wave * 16;
    for (int tj = 0; tj < 4; ++tj) {
      const int j0 = tj * 16;
      v8f acc = {};
      for (int kb = 0; kb < kD; kb += 4) {
        v2f a, b;
        a.x = s_q[i0 + m][kb + kk];
        a.y = s_q[i0 + m][kb + kk + 1];
        b.x = s_proj[kb + kk][j0 + m];
        b.y = s_proj[kb + kk + 1][j0 + m];
        acc = wmma_f32(a, b, acc);
      }
      for (int g = 0; g < 8; ++g) {
        const int row = i0 + g + (lane >> 4) * 8;
        const int col = j0 + m;
        const float val = acc[g];
        s_phi[row][col] = __sinf(val) * 0.125f;
        s_phi[row][col + kD] = __cosf(val) * 0.125f;
      }
    }
  }
  __syncthreads();

  // out_tile(128x64) = phi_q(128x128) x kv(128x64)
  {
    const int i0 = wave * 16;
    float* obase = out + (size_t)bh * kN * kD + (size_t)tile * kNTile * kD;
    for (int tj = 0; tj < 4; ++tj) {
      const int j0 = tj * 16;
      v8f acc = {};
      for (int kb = 0; kb < kE2; kb += 4) {
        v2f a, b;
        a.x = s_phi[i0 + m][kb + kk];
        a.y = s_phi[i0 + m][kb + kk + 1];
        b.x = s_kv[kb + kk][j0 + m];
        b.y = s_kv[kb + kk + 1][j0 + m];
        acc = wmma_f32(a, b, acc);
      }
      for (int g = 0; g < 8; ++g) {
        const int row = i0 + g + (lane >> 4) * 8;
        const int col = j0 + m;
        obase[row * kD + col] = acc[g];
      }
    }
  }
}

extern "C" void kernel_launch(void* const* d_in, const int* in_sizes, int n_in,
                              void* d_out, int out_size, void* d_ws, size_t ws_size,
                              hipStream_t stream) {
  (void)in_sizes; (void)n_in; (void)out_size; (void)ws_size;
  const float* q = (const float*)d_in[0];
  const float* k = (const float*)d_in[1];
  const float* v = (const float*)d_in[2];
  const float* proj = (const float*)d_in[3];
  float* out = (float*)d_out;
  float* kv_ws = (float*)d_ws;   // [BH][128][64] fp32 = 2 MB

  const int kv_elems = kBH * kE2 * kD;
  rfa_zero_kernel<<<dim3(512), dim3(256), 0, stream>>>(kv_ws, kv_elems);
  rfa_kv_kernel<<<dim3(kBH, kNSplit), dim3(256), 0, stream>>>(k, v, proj, kv_ws);
  rfa_out_kernel<<<dim3(kBH, kN / kNTile), dim3(256), 0, stream>>>(q, proj, kv_ws, out);
}